// Model_40991167873027
// MI455X (gfx1250) — compile-verified
//
#include <hip/hip_runtime.h>
#include <hip/hip_bf16.h>

// ---------------------------------------------------------------------------
// Hetero-GNN forward, algebraically restructured for MI455X (gfx1250).
//
// Key identities exploited (all exact, since the edge MLP is linear before
// the segment-mean):
//   mean_e(concat(src[s_e], ef_e) @ W + b)
//     = (sum_e src[s_e]) @ W[:64] / cnt + (sum_e ef_e) @ W[64:72] / cnt + b
// * sum_e ef_e per destination is layer-invariant -> computed once.
// * 'place'/'serve' are complete bipartite: sum_e src[s_e] = N * mean(src),
//   so those etypes reduce to one 64-vector per layer + precomputed ef means.
// * final score = op_score[o] + place_mid[o,t] + task_score[t] + b.
// Remaining per-layer work: two [4096x64]@[64x64] GEMMs (fp32 WMMA 16x16x4)
// and a 32768-edge 64-wide scatter-add. Total traffic ~25MB -> L2 resident.
// ---------------------------------------------------------------------------

#define N_OP     4096
#define N_TASK   64
#define N_TENSOR 32768
#define N_LINK   1024
#define N_PLACE  (N_OP * N_TASK)
#define NODE_H   64
#define EDGE_H   8
#define RAW_OP   256
#define RAW_TASK 64
#define RAW_E    16
#define N_LAYERS 6
#define GK       (NODE_H + EDGE_H)   // 72

typedef float v2f __attribute__((ext_vector_type(2)));
typedef float v8f __attribute__((ext_vector_type(8)));

__device__ __forceinline__ float eluf(float x) {
    return x > 0.f ? x : (expf(x) - 1.f);
}

// One wave computes a 16-row x 64-col fp32 GEMM tile with V_WMMA_F32_16X16X4_F32.
// A: row-major [16 x K] (lda = K step arbitrary), B: row-major [K x 64].
// ISA 16x4 f32 A layout: lanes 0-15 hold M=lane {K=k,k+1}; lanes 16-31 hold
// M=lane-16 {K=k+2,k+3}. B assumed mirrored (N=lane%16, same K striping).
__device__ __forceinline__ void wave_gemm_16x64(const float* __restrict__ A, int lda,
                                                const float* __restrict__ B, int K,
                                                v8f c[4]) {
    const int lane = threadIdx.x & 31;
    const int half = lane >> 4;
    const int l16  = lane & 15;
    const float* arow = A + l16 * lda;
    for (int k = 0; k < K; k += 4) {
        const int ka = k + half * 2;
        v2f a;
        a[0] = arow[ka];
        a[1] = arow[ka + 1];
#pragma unroll
        for (int nt = 0; nt < 4; ++nt) {
            const int n = nt * 16 + l16;
            v2f b;
            b[0] = B[ka * NODE_H + n];
            b[1] = B[(ka + 1) * NODE_H + n];
            c[nt] = __builtin_amdgcn_wmma_f32_16x16x4_f32(
                false, a, false, b, (short)0, c[nt], false, false);
        }
    }
}

// ---- node feature projection: Y = elu(X @ W + b), X:[tiles*16 x K] ----------
__global__ void k_proj(const float* __restrict__ X, const float* __restrict__ W,
                       const float* __restrict__ b, float* __restrict__ Y, int K) {
    const int wave = threadIdx.x >> 5;
    const int tile = blockIdx.x * 4 + wave;
    const int lane = threadIdx.x & 31;
    const int half = lane >> 4, l16 = lane & 15;
    v8f c[4] = {};
    wave_gemm_16x64(X + (size_t)tile * 16 * K, K, W, K, c);
#pragma unroll
    for (int nt = 0; nt < 4; ++nt) {
        const int n = nt * 16 + l16;
#pragma unroll
        for (int j = 0; j < 8; ++j) {
            const int row = tile * 16 + j + 8 * half;
            Y[row * NODE_H + n] = eluf(c[nt][j] + b[n]);
        }
    }
}

// ---- per-edge elu(feat@eW+eb) aggregated by destination (layer-invariant) --
__global__ void k_edge_agg(const float* __restrict__ feats, int nE,
                           const int* __restrict__ dstA, const int* __restrict__ dstB,
                           const float* __restrict__ eW, const float* __restrict__ eb,
                           int iA, int iB,
                           float* __restrict__ aggA, float* __restrict__ cntA,
                           float* __restrict__ aggB, float* __restrict__ cntB) {
    const int e = blockIdx.x * blockDim.x + threadIdx.x;
    if (e >= nE) return;
    float f[RAW_E];
    const float* fp = feats + (size_t)e * RAW_E;
#pragma unroll
    for (int k = 0; k < RAW_E; ++k) f[k] = fp[k];

    const int dA = dstA[e];
    const float* WA = eW + iA * RAW_E * EDGE_H;
    const float* bA = eb + iA * EDGE_H;
#pragma unroll
    for (int j = 0; j < EDGE_H; ++j) {
        float s = bA[j];
#pragma unroll
        for (int k = 0; k < RAW_E; ++k) s += f[k] * WA[k * EDGE_H + j];
        atomicAdd(&aggA[dA * EDGE_H + j], eluf(s));
    }
    atomicAdd(&cntA[dA], 1.f);

    if (aggB) {
        const int dB = dstB[e];
        const float* WB = eW + iB * RAW_E * EDGE_H;
        const float* bB = eb + iB * EDGE_H;
#pragma unroll
        for (int j = 0; j < EDGE_H; ++j) {
            float s = bB[j];
#pragma unroll
            for (int k = 0; k < RAW_E; ++k) s += f[k] * WB[k * EDGE_H + j];
            atomicAdd(&aggB[dB * EDGE_H + j], eluf(s));
        }
        atomicAdd(&cntB[dB], 1.f);
    }
}

// ---- single pass over place_feats (16MB): ef_place/ef_serve, their means,
//      and the per-place-edge final-W middle term ---------------------------
__global__ void k_place_serve(const float* __restrict__ pf,
                              const float* __restrict__ eW, const float* __restrict__ eb,
                              const float* __restrict__ fW,
                              float* __restrict__ place_mid,
                              float* __restrict__ mserve,   // [N_OP][8], mean over tasks
                              float* __restrict__ mplace) { // [N_TASK][8], sums (scaled later)
    __shared__ float red[256][EDGE_H];
    __shared__ float psum[N_TASK * EDGE_H];
    const int tid = threadIdx.x;
    for (int i = tid; i < N_TASK * EDGE_H; i += 256) psum[i] = 0.f;
    __syncthreads();

    const int g = tid >> 6;            // 0..3 : op within block
    const int t = tid & 63;            // task
    const int o = blockIdx.x * 4 + g;
    const int e = o * N_TASK + t;

    float f[RAW_E];
    const float* fp = pf + (size_t)e * RAW_E;
#pragma unroll
    for (int k = 0; k < RAW_E; ++k) f[k] = fp[k];

    const float* Wp = eW + 3 * RAW_E * EDGE_H;
    const float* bp = eb + 3 * EDGE_H;
    const float* Ws = eW + 4 * RAW_E * EDGE_H;
    const float* bs = eb + 4 * EDGE_H;
    float efp[EDGE_H], efs[EDGE_H];
#pragma unroll
    for (int j = 0; j < EDGE_H; ++j) {
        float sp = bp[j], ss = bs[j];
#pragma unroll
        for (int k = 0; k < RAW_E; ++k) {
            sp += f[k] * Wp[k * EDGE_H + j];
            ss += f[k] * Ws[k * EDGE_H + j];
        }
        efp[j] = eluf(sp);
        efs[j] = eluf(ss);
    }

    float pm = 0.f;
#pragma unroll
    for (int j = 0; j < EDGE_H; ++j) pm += efp[j] * fW[NODE_H + j];
    place_mid[e] = pm;

    // serve: mean over the 64 tasks of this op (LDS tree within 64-thread group)
#pragma unroll
    for (int j = 0; j < EDGE_H; ++j) red[tid][j] = efs[j];
    __syncthreads();
    for (int s = 32; s >= 1; s >>= 1) {
        if (t < s) {
#pragma unroll
            for (int j = 0; j < EDGE_H; ++j) red[tid][j] += red[tid + s][j];
        }
        __syncthreads();
    }
    if (t < EDGE_H) mserve[o * EDGE_H + t] = red[g * 64][t] * (1.f / N_TASK);

    // place: per-task column sums, block partial in LDS then one global atomic
#pragma unroll
    for (int j = 0; j < EDGE_H; ++j) atomicAdd(&psum[t * EDGE_H + j], efp[j]);
    __syncthreads();
    for (int i = tid; i < N_TASK * EDGE_H; i += 256) atomicAdd(&mplace[i], psum[i]);
}

__global__ void k_scale(float* __restrict__ p, int n, float f) {
    const int i = blockIdx.x * blockDim.x + threadIdx.x;
    if (i < n) p[i] *= f;
}

// ---- per-layer prep: node means and the broadcast base vectors -------------
__global__ void k_layer_prep(const float* __restrict__ opc, const float* __restrict__ taskc,
                             const float* __restrict__ gW, const float* __restrict__ gb,
                             int l, float* __restrict__ place_base,
                             float* __restrict__ serve_base) {
    __shared__ float part[256];
    __shared__ float om[64], tm[64];
    const int tid = threadIdx.x;
    const int n = tid & 63, chunk = tid >> 6;
    float s = 0.f;
    const int r0 = chunk * (N_OP / 4), r1 = r0 + N_OP / 4;
    for (int r = r0; r < r1; ++r) s += opc[r * NODE_H + n];
    part[tid] = s;
    __syncthreads();
    if (tid < 64) {
        om[tid] = (part[tid] + part[64 + tid] + part[128 + tid] + part[192 + tid]) * (1.f / N_OP);
        float ts = 0.f;
        for (int r = 0; r < N_TASK; ++r) ts += taskc[r * NODE_H + tid];
        tm[tid] = ts * (1.f / N_TASK);
    }
    __syncthreads();
    if (tid < 64) {
        const float* W3 = gW + (size_t)(3 * N_LAYERS + l) * GK * NODE_H;
        const float* W4 = gW + (size_t)(4 * N_LAYERS + l) * GK * NODE_H;
        const float* b3 = gb + (3 * N_LAYERS + l) * NODE_H;
        const float* b4 = gb + (4 * N_LAYERS + l) * NODE_H;
        float pb = b3[tid], sb = b4[tid];
        for (int k = 0; k < NODE_H; ++k) {
            pb += om[k] * W3[k * NODE_H + tid];
            sb += tm[k] * W4[k * NODE_H + tid];
        }
        place_base[tid] = pb;   // place msg base (dst=task), cnt = N_OP
        serve_base[tid] = sb;   // serve msg base (dst=op),   cnt = N_TASK
    }
}

// ---- scatter-add of source node vectors (prev + succ share the edge list) --
__global__ void k_scatter_op(const float* __restrict__ opc,
                             const int* __restrict__ psrc, const int* __restrict__ pdst,
                             float* __restrict__ Sprev, float* __restrict__ Ssucc) {
    const int gid = blockIdx.x * blockDim.x + threadIdx.x;
    const int e = gid >> 6, n = gid & 63;
    if (e >= N_TENSOR) return;
    const int s = psrc[e], d = pdst[e];
    atomicAdd(&Sprev[d * NODE_H + n], opc[s * NODE_H + n]);
    atomicAdd(&Ssucc[s * NODE_H + n], opc[d * NODE_H + n]);
}

__global__ void k_scatter_link(const float* __restrict__ taskc,
                               const int* __restrict__ lsrc, const int* __restrict__ ldst,
                               float* __restrict__ Slink) {
    const int gid = blockIdx.x * blockDim.x + threadIdx.x;
    const int e = gid >> 6, n = gid & 63;
    if (e >= N_LINK) return;
    atomicAdd(&Slink[ldst[e] * NODE_H + n], taskc[lsrc[e] * NODE_H + n]);
}

// ---- op update: prev/succ WMMA GEMMs + serve broadcast, residual + elu -----
__global__ void k_op_layer(const float* __restrict__ opc, float* __restrict__ opn,
                           const float* __restrict__ Sprev, const float* __restrict__ Ssucc,
                           const float* __restrict__ aggp, const float* __restrict__ cntp,
                           const float* __restrict__ aggs, const float* __restrict__ cnts,
                           const float* __restrict__ mserve, const float* __restrict__ serve_base,
                           const float* __restrict__ gW, const float* __restrict__ gb, int l) {
    const int wave = threadIdx.x >> 5;
    const int tile = blockIdx.x * 4 + wave;
    const int lane = threadIdx.x & 31;
    const int half = lane >> 4, l16 = lane & 15;
    const float* W1 = gW + (size_t)(1 * N_LAYERS + l) * GK * NODE_H;
    const float* W2 = gW + (size_t)(2 * N_LAYERS + l) * GK * NODE_H;
    const float* W4 = gW + (size_t)(4 * N_LAYERS + l) * GK * NODE_H;
    const float* b1 = gb + (1 * N_LAYERS + l) * NODE_H;
    const float* b2 = gb + (2 * N_LAYERS + l) * NODE_H;
    v8f cp[4] = {}, cs[4] = {};
    wave_gemm_16x64(Sprev + (size_t)tile * 16 * NODE_H, NODE_H, W1, NODE_H, cp);
    wave_gemm_16x64(Ssucc + (size_t)tile * 16 * NODE_H, NODE_H, W2, NODE_H, cs);
    const float* W1b = W1 + NODE_H * NODE_H;   // rows 64..71 (edge-feature slice)
    const float* W2b = W2 + NODE_H * NODE_H;
    const float* W4b = W4 + NODE_H * NODE_H;
#pragma unroll
    for (int j = 0; j < 8; ++j) {
        const int row = tile * 16 + j + 8 * half;
        const float cpc = cntp[row], csc = cnts[row];
        const float rcp = 1.f / fmaxf(cpc, 1.f), rcs = 1.f / fmaxf(csc, 1.f);
        float ap[8], as_[8], ms[8];
#pragma unroll
        for (int q = 0; q < 8; ++q) {
            ap[q]  = aggp[row * 8 + q];
            as_[q] = aggs[row * 8 + q];
            ms[q]  = mserve[row * 8 + q];
        }
#pragma unroll
        for (int nt = 0; nt < 4; ++nt) {
            const int n = nt * 16 + l16;
            float dp = 0.f, ds = 0.f, dv = 0.f;
#pragma unroll
            for (int q = 0; q < 8; ++q) {
                dp += ap[q]  * W1b[q * NODE_H + n];
                ds += as_[q] * W2b[q * NODE_H + n];
                dv += ms[q]  * W4b[q * NODE_H + n];
            }
            const float vp = (cp[nt][j] + dp + cpc * b1[n]) * rcp;
            const float vs = (cs[nt][j] + ds + csc * b2[n]) * rcs;
            const float vv = serve_base[n] + dv;
            const float x = opc[row * NODE_H + n] + (vp + vs + vv) * (1.f / 3.f);
            opn[row * NODE_H + n] = eluf(x);
        }
    }
}

// ---- task update: link WMMA GEMM + place broadcast -------------------------
__global__ void k_task_layer(const float* __restrict__ taskc, float* __restrict__ taskn,
                             const float* __restrict__ Slink,
                             const float* __restrict__ aggl, const float* __restrict__ cntl,
                             const float* __restrict__ mplace, const float* __restrict__ place_base,
                             const float* __restrict__ gW, const float* __restrict__ gb, int l) {
    const int wave = threadIdx.x >> 5;   // 4 waves, 16 rows each = 64 tasks
    const int lane = threadIdx.x & 31;
    const int half = lane >> 4, l16 = lane & 15;
    const float* W0 = gW + (size_t)(0 * N_LAYERS + l) * GK * NODE_H;
    const float* W3 = gW + (size_t)(3 * N_LAYERS + l) * GK * NODE_H;
    const float* b0 = gb + (0 * N_LAYERS + l) * NODE_H;
    v8f c[4] = {};
    wave_gemm_16x64(Slink + (size_t)wave * 16 * NODE_H, NODE_H, W0, NODE_H, c);
    const float* W0b = W0 + NODE_H * NODE_H;
    const float* W3b = W3 + NODE_H * NODE_H;
#pragma unroll
    for (int j = 0; j < 8; ++j) {
        const int row = wave * 16 + j + 8 * half;
        const float cl = cntl[row];
        const float rcl = 1.f / fmaxf(cl, 1.f);
        float al[8], mp[8];
#pragma unroll
        for (int q = 0; q < 8; ++q) {
            al[q] = aggl[row * 8 + q];
            mp[q] = mplace[row * 8 + q];
        }
#pragma unroll
        for (int nt = 0; nt < 4; ++nt) {
            const int n = nt * 16 + l16;
            float dl = 0.f, dv = 0.f;
#pragma unroll
            for (int q = 0; q < 8; ++q) {
                dl += al[q] * W0b[q * NODE_H + n];
                dv += mp[q] * W3b[q * NODE_H + n];
            }
            const float vl = (c[nt][j] + dl + cl * b0[n]) * rcl;
            const float vp = place_base[n] + dv;
            const float x = taskc[row * NODE_H + n] + (vl + vp) * 0.5f;
            taskn[row * NODE_H + n] = eluf(x);
        }
    }
}

// ---- final decomposition ---------------------------------------------------
__global__ void k_scores(const float* __restrict__ opf, const float* __restrict__ taskf,
                         const float* __restrict__ fW,
                         float* __restrict__ op_score, float* __restrict__ task_score) {
    const int gid = blockIdx.x * blockDim.x + threadIdx.x;
    if (gid < N_OP) {
        float s = 0.f;
        for (int k = 0; k < NODE_H; ++k) s += opf[gid * NODE_H + k] * fW[k];
        op_score[gid] = s;
    } else if (gid < N_OP + N_TASK) {
        const int t = gid - N_OP;
        float s = 0.f;
        for (int k = 0; k < NODE_H; ++k) s += taskf[t * NODE_H + k] * fW[GK + k];
        task_score[t] = s;
    }
}

__global__ void k_final(const float* __restrict__ op_score, const float* __restrict__ task_score,
                        const float* __restrict__ place_mid, const float* __restrict__ fb,
                        float* __restrict__ out) {
    const int e = blockIdx.x * blockDim.x + threadIdx.x;
    if (e >= N_PLACE) return;
    out[e] = op_score[e >> 6] + place_mid[e] + task_score[e & 63] + fb[0];
}

// ---------------------------------------------------------------------------
extern "C" void kernel_launch(void* const* d_in, const int* in_sizes, int n_in,
                              void* d_out, int out_size, void* d_ws, size_t ws_size,
                              hipStream_t stream) {
    const float* op_feats     = (const float*)d_in[0];
    const float* task_feats   = (const float*)d_in[1];
    const float* tensor_feats = (const float*)d_in[2];
    const float* link_feats   = (const float*)d_in[3];
    const float* place_feats  = (const float*)d_in[4];
    const int*   prev_src     = (const int*)d_in[5];
    const int*   prev_dst     = (const int*)d_in[6];
    const int*   link_src     = (const int*)d_in[7];
    const int*   link_dst     = (const int*)d_in[8];
    const float* op_W   = (const float*)d_in[9];
    const float* op_b   = (const float*)d_in[10];
    const float* task_W = (const float*)d_in[11];
    const float* task_b = (const float*)d_in[12];
    const float* eW     = (const float*)d_in[13];
    const float* eb     = (const float*)d_in[14];
    const float* gW     = (const float*)d_in[15];
    const float* gb     = (const float*)d_in[16];
    const float* fW     = (const float*)d_in[17];
    const float* fb     = (const float*)d_in[18];

    float* ws = (float*)d_ws;
    size_t off = 0;
    auto alloc = [&](size_t n) { float* p = ws + off; off += n; return p; };
    float* opA    = alloc((size_t)N_OP * NODE_H);
    float* opB    = alloc((size_t)N_OP * NODE_H);
    float* taskA  = alloc((size_t)N_TASK * NODE_H);
    float* taskB  = alloc((size_t)N_TASK * NODE_H);
    float* Sprev  = alloc((size_t)N_OP * NODE_H);   // contiguous zero region A
    float* Ssucc  = alloc((size_t)N_OP * NODE_H);
    float* Slink  = alloc((size_t)N_TASK * NODE_H);
    float* zb     = ws + off;                       // contiguous zero region B
    float* aggp   = alloc((size_t)N_OP * EDGE_H);
    float* aggs   = alloc((size_t)N_OP * EDGE_H);
    float* aggl   = alloc((size_t)N_TASK * EDGE_H);
    float* cntp   = alloc(N_OP);
    float* cnts   = alloc(N_OP);
    float* cntl   = alloc(N_TASK);
    float* mplace = alloc((size_t)N_TASK * EDGE_H);
    size_t zb_n   = (size_t)(ws + off - zb);
    float* mserve    = alloc((size_t)N_OP * EDGE_H);  // fully written, no zero needed
    float* place_mid = alloc((size_t)N_PLACE);
    float* place_base = alloc(64);
    float* serve_base = alloc(64);
    float* op_score   = alloc(N_OP);
    float* task_score = alloc(N_TASK);
    if (ws_size < off * sizeof(float)) return;

    // per-launch zeroing of atomic accumulators (graph-capture safe)
    hipMemsetAsync(zb, 0, zb_n * sizeof(float), stream);

    // ---- layer-invariant precompute ----
    k_proj<<<N_OP / 64, 128, 0, stream>>>(op_feats, op_W, op_b, opA, RAW_OP);
    k_proj<<<1, 128, 0, stream>>>(task_feats, task_W, task_b, taskA, RAW_TASK);
    k_edge_agg<<<(N_TENSOR + 255) / 256, 256, 0, stream>>>(
        tensor_feats, N_TENSOR, prev_dst, prev_src, eW, eb, 1, 2,
        aggp, cntp, aggs, cnts);
    k_edge_agg<<<(N_LINK + 255) / 256, 256, 0, stream>>>(
        link_feats, N_LINK, link_dst, link_dst, eW, eb, 0, 0,
        aggl, cntl, nullptr, nullptr);
    k_place_serve<<<N_OP / 4, 256, 0, stream>>>(place_feats, eW, eb, fW,
                                                place_mid, mserve, mplace);
    k_scale<<<2, 256, 0, stream>>>(mplace, N_TASK * EDGE_H, 1.f / N_OP);

    // ---- 6 message-passing layers (double-buffered) ----
    float* cur_op = opA; float* nxt_op = opB;
    float* cur_task = taskA; float* nxt_task = taskB;
    for (int l = 0; l < N_LAYERS; ++l) {
        k_layer_prep<<<1, 256, 0, stream>>>(cur_op, cur_task, gW, gb, l,
                                            place_base, serve_base);
        hipMemsetAsync(Sprev, 0,
                       ((size_t)N_OP * NODE_H * 2 + (size_t)N_TASK * NODE_H) * sizeof(float),
                       stream);
        k_scatter_op<<<(N_TENSOR * 64) / 256, 256, 0, stream>>>(
            cur_op, prev_src, prev_dst, Sprev, Ssucc);
        k_scatter_link<<<(N_LINK * 64) / 256, 256, 0, stream>>>(
            cur_task, link_src, link_dst, Slink);
        k_op_layer<<<N_OP / 64, 128, 0, stream>>>(
            cur_op, nxt_op, Sprev, Ssucc, aggp, cntp, aggs, cnts,
            mserve, serve_base, gW, gb, l);
        k_task_layer<<<1, 128, 0, stream>>>(
            cur_task, nxt_task, Slink, aggl, cntl, mplace, place_base, gW, gb, l);
        float* t;
        t = cur_op; cur_op = nxt_op; nxt_op = t;
        t = cur_task; cur_task = nxt_task; nxt_task = t;
    }

    // ---- final score ----
    k_scores<<<(N_OP + N_TASK + 255) / 256, 256, 0, stream>>>(
        cur_op, cur_task, fW, op_score, task_score);
    k_final<<<N_PLACE / 256, 256, 0, stream>>>(
        op_score, task_score, place_mid, fb, (float*)d_out);
}